// SimpleGAT_60997125538490
// MI455X (gfx1250) — compile-verified
//
#include <hip/hip_runtime.h>
#include <math.h>

// ---------------------------------------------------------------------------
// GATv2 x3 + GraphNorm + mean-pool + linear for MI455X (gfx1250, wave32).
// GEMMs (x@Wl, x@Wr) run on v_wmma_f32_16x16x4_f32 (exact fp32).
// Edge gather/softmax/scatter passes exploit the 192MB L2 (node features are
// L2-resident), using int max/min atomics for segment max, f32 atomic adds for sums.
// ---------------------------------------------------------------------------

typedef float v2f __attribute__((ext_vector_type(2)));
typedef float v8f __attribute__((ext_vector_type(8)));

#define N_NODES 100000
#define N_EDGES 1600000
#define N_GRAPH 100
#define ETOT    (N_NODES + N_EDGES)   // self-loops appended implicitly
#define NEG_SLOPE 0.2f
#define EPS_GN    1e-5f

// float atomic max via sign-magnitude int trick (correct for mixed signs)
__device__ inline void atomicMaxF(float* addr, float v) {
  if (v >= 0.0f) atomicMax((int*)addr, __float_as_int(v));
  else           atomicMin((unsigned int*)addr, __float_as_uint(v));
}

__global__ void fill_kernel(float* __restrict__ p, float v, int n) {
  int i = blockIdx.x * blockDim.x + threadIdx.x;
  if (i < n) p[i] = v;
}

__global__ void count_kernel(const long long* __restrict__ batch,
                             float* __restrict__ cnt) {
  int i = blockIdx.x * blockDim.x + threadIdx.x;
  if (i < N_NODES) atomicAdd(&cnt[(int)batch[i]], 1.0f);
}

// ---------------------------------------------------------------------------
// WMMA GEMM: Out[M,Dout] = A[M,64] @ W[64,Dout], one 16x16 tile per wave.
// Per CDNA5 ISA f32 layouts:
//   A 16x4 : lanes 0-15 row M=lane, VGPR0/1 = K0/K1; lanes 16-31 = K2/K3
//   B 4x16 : lanes 0-15 col N=lane, VGPR0/1 = K0/K1; lanes 16-31 = K2/K3
//   C/D    : lane%16 = N, VGPR r -> M = r (+8 for lanes 16-31)
// ---------------------------------------------------------------------------
__global__ __launch_bounds__(256) void gemm_wmma(const float* __restrict__ A,
                                                 const float* __restrict__ W,
                                                 float* __restrict__ Out,
                                                 int M, int Dout) {
  const int lane = threadIdx.x & 31;
  const int wave = threadIdx.x >> 5;
  const int ntn  = Dout >> 4;
  const int tile = blockIdx.x * (blockDim.x >> 5) + wave;
  if (tile >= (M >> 4) * ntn) return;            // whole-wave exit: EXEC all-1s
  const int m0   = (tile / ntn) << 4;
  const int n0   = (tile % ntn) << 4;
  const int half = lane >> 4;                    // 0: K0/K1, 1: K2/K3
  const int l16  = lane & 15;
  const int arow = m0 + l16;
  const int bcol = n0 + l16;
  v8f acc = {};
#pragma unroll
  for (int k = 0; k < 64; k += 4) {
    const int kk = k + (half << 1);
    v2f a, b;
    a.x = A[arow * 64 + kk];
    a.y = A[arow * 64 + kk + 1];
    b.x = W[kk * Dout + bcol];
    b.y = W[(kk + 1) * Dout + bcol];
    acc = __builtin_amdgcn_wmma_f32_16x16x4_f32(false, a, false, b,
                                                (short)0, acc, false, false);
  }
  const int orow = m0 + (half << 3);
#pragma unroll
  for (int r = 0; r < 8; ++r)
    Out[(orow + r) * Dout + bcol] = acc[r];
}

// ---------------------------------------------------------------------------
// Edge pass A: e[e,h] = sum_c att[h,c]*leaky_relu(xl[src]+xr[dst]); seg-max
// ---------------------------------------------------------------------------
__global__ void edge_scores(const long long* __restrict__ ei,
                            const float* __restrict__ xl,
                            const float* __restrict__ xr,
                            const float* __restrict__ att,
                            float* __restrict__ ebuf,
                            float* __restrict__ emax,
                            int D, int C) {
  int tid = blockIdx.x * blockDim.x + threadIdx.x;
  if (tid >= ETOT * 4) return;
  int e = tid >> 2, h = tid & 3;
  int src, dst;
  if (e < N_EDGES) { src = (int)ei[e]; dst = (int)ei[N_EDGES + e]; }
  else             { src = e - N_EDGES; dst = src; }
  const float4* xls = (const float4*)(xl + (size_t)src * D + h * C);
  const float4* xrs = (const float4*)(xr + (size_t)dst * D + h * C);
  const float4* ah  = (const float4*)(att + h * C);
  float s = 0.0f;
  for (int j = 0; j < (C >> 2); ++j) {
    float4 a = xls[j], b = xrs[j], w = ah[j];
    float m0 = a.x + b.x; m0 = m0 >= 0.f ? m0 : NEG_SLOPE * m0;
    float m1 = a.y + b.y; m1 = m1 >= 0.f ? m1 : NEG_SLOPE * m1;
    float m2 = a.z + b.z; m2 = m2 >= 0.f ? m2 : NEG_SLOPE * m2;
    float m3 = a.w + b.w; m3 = m3 >= 0.f ? m3 : NEG_SLOPE * m3;
    s += w.x * m0 + w.y * m1 + w.z * m2 + w.w * m3;
  }
  ebuf[tid] = s;
  atomicMaxF(&emax[dst * 4 + h], s);
}

// Edge pass B: ex = exp(e - emax[dst]); seg-sum denom
__global__ void edge_exp(const long long* __restrict__ ei,
                         float* __restrict__ ebuf,
                         const float* __restrict__ emax,
                         float* __restrict__ den) {
  int tid = blockIdx.x * blockDim.x + threadIdx.x;
  if (tid >= ETOT * 4) return;
  int e = tid >> 2, h = tid & 3;
  int dst = (e < N_EDGES) ? (int)ei[N_EDGES + e] : (e - N_EDGES);
  float ex = expf(ebuf[tid] - emax[dst * 4 + h]);
  ebuf[tid] = ex;
  atomicAdd(&den[dst * 4 + h], ex);
}

// Edge pass C: agg[dst] += (ex/den[dst]) * xl[src]
__global__ void edge_agg(const long long* __restrict__ ei,
                         const float* __restrict__ ebuf,
                         const float* __restrict__ den,
                         const float* __restrict__ xl,
                         float* __restrict__ agg,
                         int D, int C) {
  int tid = blockIdx.x * blockDim.x + threadIdx.x;
  if (tid >= ETOT * 4) return;
  int e = tid >> 2, h = tid & 3;
  int src, dst;
  if (e < N_EDGES) { src = (int)ei[e]; dst = (int)ei[N_EDGES + e]; }
  else             { src = e - N_EDGES; dst = src; }
  float alpha = ebuf[tid] / den[dst * 4 + h];
  const float* xs = xl + (size_t)src * D + h * C;
  float* o = agg + (size_t)dst * D + h * C;
  for (int c = 0; c < C; ++c) atomicAdd(&o[c], alpha * xs[c]);
}

// bias add (+ optional head-mean for layer 3) and per-graph feature sum
__global__ void finalize_kernel(const float* __restrict__ agg,
                                const float* __restrict__ bias,
                                const long long* __restrict__ batch,
                                float* __restrict__ xn,
                                float* __restrict__ gsum,
                                int D, int headMean) {
  int tid = blockIdx.x * blockDim.x + threadIdx.x;
  if (tid >= N_NODES * D) return;
  int n = tid / D, d = tid - n * D;
  float t;
  if (headMean) {
    const float* a = agg + (size_t)n * 128 + d;      // D==32, H==4, C==32
    t = 0.25f * (a[0] + a[32] + a[64] + a[96]) + bias[d];
  } else {
    t = agg[tid] + bias[d];
  }
  xn[tid] = t;
  atomicAdd(&gsum[(int)batch[n] * D + d], t);
}

__global__ void gn_mean(float* __restrict__ gsum, const float* __restrict__ cnt,
                        int D) {
  int tid = blockIdx.x * blockDim.x + threadIdx.x;
  if (tid >= N_GRAPH * D) return;
  gsum[tid] /= fmaxf(cnt[tid / D], 1.0f);
}

__global__ void gn_center(float* __restrict__ xn,
                          const float* __restrict__ gmean,
                          const float* __restrict__ ga,
                          const long long* __restrict__ batch,
                          float* __restrict__ gvar, int D) {
  int tid = blockIdx.x * blockDim.x + threadIdx.x;
  if (tid >= N_NODES * D) return;
  int n = tid / D, d = tid - n * D;
  int g = (int)batch[n];
  float xc = xn[tid] - ga[d] * gmean[g * D + d];
  xn[tid] = xc;
  atomicAdd(&gvar[g * D + d], xc * xc);
}

__global__ void gn_rstd(float* __restrict__ gvar, const float* __restrict__ cnt,
                        int D) {
  int tid = blockIdx.x * blockDim.x + threadIdx.x;
  if (tid >= N_GRAPH * D) return;
  gvar[tid] = rsqrtf(gvar[tid] / fmaxf(cnt[tid / D], 1.0f) + EPS_GN);
}

__global__ void gn_out(const float* __restrict__ xn,
                       const float* __restrict__ grstd,
                       const float* __restrict__ gw,
                       const float* __restrict__ gb,
                       const long long* __restrict__ batch,
                       float* __restrict__ outp, int D) {
  int tid = blockIdx.x * blockDim.x + threadIdx.x;
  if (tid >= N_NODES * D) return;
  int n = tid / D, d = tid - n * D;
  int g = (int)batch[n];
  float y = gw[d] * xn[tid] * grstd[g * D + d] + gb[d];
  outp[tid] = fmaxf(y, 0.0f);   // fused ReLU
}

__global__ void pool_kernel(const float* __restrict__ h3,
                            const long long* __restrict__ batch,
                            float* __restrict__ pool) {
  int tid = blockIdx.x * blockDim.x + threadIdx.x;
  if (tid >= N_NODES * 32) return;
  int n = tid >> 5, c = tid & 31;
  atomicAdd(&pool[(int)batch[n] * 32 + c], h3[tid]);
}

__global__ void final_kernel(const float* __restrict__ pool,
                             const float* __restrict__ cnt,
                             const float* __restrict__ Wlin,
                             const float* __restrict__ blin,
                             float* __restrict__ outp) {
  int tid = blockIdx.x * blockDim.x + threadIdx.x;
  if (tid >= N_GRAPH * 2) return;
  int g = tid >> 1, j = tid & 1;
  float ic = 1.0f / fmaxf(cnt[g], 1.0f);
  float s = blin[j];
  for (int c = 0; c < 32; ++c) s += pool[g * 32 + c] * ic * Wlin[c * 2 + j];
  outp[tid] = s;
}

// ---------------------------------------------------------------------------
extern "C" void kernel_launch(void* const* d_in, const int* in_sizes, int n_in,
                              void* d_out, int out_size, void* d_ws,
                              size_t ws_size, hipStream_t stream) {
  (void)in_sizes; (void)n_in; (void)out_size; (void)ws_size;
  const float*     x     = (const float*)d_in[0];
  const long long* ei    = (const long long*)d_in[1];
  const long long* batch = (const long long*)d_in[2];
  const float* Wl1 = (const float*)d_in[3],  *Wr1 = (const float*)d_in[4];
  const float* att1= (const float*)d_in[5],  *b1  = (const float*)d_in[6];
  const float* g1w = (const float*)d_in[7],  *g1b = (const float*)d_in[8];
  const float* g1a = (const float*)d_in[9];
  const float* Wl2 = (const float*)d_in[10], *Wr2 = (const float*)d_in[11];
  const float* att2= (const float*)d_in[12], *b2  = (const float*)d_in[13];
  const float* g2w = (const float*)d_in[14], *g2b = (const float*)d_in[15];
  const float* g2a = (const float*)d_in[16];
  const float* Wl3 = (const float*)d_in[17], *Wr3 = (const float*)d_in[18];
  const float* att3= (const float*)d_in[19], *b3  = (const float*)d_in[20];
  const float* g3w = (const float*)d_in[21], *g3b = (const float*)d_in[22];
  const float* g3a = (const float*)d_in[23];
  const float* Wlin= (const float*)d_in[24], *blin= (const float*)d_in[25];
  float* out = (float*)d_out;

  // workspace layout (floats); total ~236 MB
  float* ws = (float*)d_ws;
  size_t o = 0;
  float* xl   = ws + o; o += (size_t)N_NODES * 128;
  float* xr   = ws + o; o += (size_t)N_NODES * 128;   // reused as xn post-edges
  float* agg  = ws + o; o += (size_t)N_NODES * 128;
  float* ebuf = ws + o; o += (size_t)ETOT * 4;
  float* emax = ws + o; o += (size_t)N_NODES * 4;
  float* den  = ws + o; o += (size_t)N_NODES * 4;
  float* hA   = ws + o; o += (size_t)N_NODES * 64;
  float* hB   = ws + o; o += (size_t)N_NODES * 64;
  float* cnt  = ws + o; o += N_GRAPH;
  float* gsum = ws + o; o += N_GRAPH * 128;
  float* gvar = ws + o; o += N_GRAPH * 128;
  float* pool = ws + o; o += N_GRAPH * 32;

  auto FILL = [&](float* p, float v, size_t n) {
    fill_kernel<<<(int)((n + 255) / 256), 256, 0, stream>>>(p, v, (int)n);
  };

  FILL(cnt, 0.0f, N_GRAPH);
  count_kernel<<<(N_NODES + 255) / 256, 256, 0, stream>>>(batch, cnt);

  auto layer = [&](const float* xin, const float* Wl, const float* Wr,
                   const float* att, const float* bias, const float* gw,
                   const float* gb, const float* ga, int Dout, int C, int Dn,
                   int headMean, float* hout) {
    const int tiles = (N_NODES / 16) * (Dout / 16);
    const int gblk  = (tiles + 7) / 8;                 // 8 waves / block
    gemm_wmma<<<gblk, 256, 0, stream>>>(xin, Wl, xl, N_NODES, Dout);
    gemm_wmma<<<gblk, 256, 0, stream>>>(xin, Wr, xr, N_NODES, Dout);
    FILL(emax, -INFINITY, (size_t)N_NODES * 4);
    FILL(den, 0.0f, (size_t)N_NODES * 4);
    FILL(agg, 0.0f, (size_t)N_NODES * Dout);
    FILL(gsum, 0.0f, (size_t)N_GRAPH * Dn);
    FILL(gvar, 0.0f, (size_t)N_GRAPH * Dn);
    const int eth = ETOT * 4, eblk = (eth + 255) / 256;
    edge_scores<<<eblk, 256, 0, stream>>>(ei, xl, xr, att, ebuf, emax, Dout, C);
    edge_exp<<<eblk, 256, 0, stream>>>(ei, ebuf, emax, den);
    edge_agg<<<eblk, 256, 0, stream>>>(ei, ebuf, den, xl, agg, Dout, C);
    const int nth = N_NODES * Dn, nblk = (nth + 255) / 256;
    const int gth = N_GRAPH * Dn, gbk = (gth + 255) / 256;
    finalize_kernel<<<nblk, 256, 0, stream>>>(agg, bias, batch, xr, gsum, Dn,
                                              headMean);
    gn_mean<<<gbk, 256, 0, stream>>>(gsum, cnt, Dn);
    gn_center<<<nblk, 256, 0, stream>>>(xr, gsum, ga, batch, gvar, Dn);
    gn_rstd<<<gbk, 256, 0, stream>>>(gvar, cnt, Dn);
    gn_out<<<nblk, 256, 0, stream>>>(xr, gvar, gw, gb, batch, hout, Dn);
  };

  layer(x,  Wl1, Wr1, att1, b1, g1w, g1b, g1a,  64, 16, 64, 0, hA);
  layer(hA, Wl2, Wr2, att2, b2, g2w, g2b, g2a,  64, 16, 64, 0, hB);
  layer(hB, Wl3, Wr3, att3, b3, g3w, g3b, g3a, 128, 32, 32, 1, hA);

  FILL(pool, 0.0f, (size_t)N_GRAPH * 32);
  pool_kernel<<<(N_NODES * 32 + 255) / 256, 256, 0, stream>>>(hA, batch, pool);
  final_kernel<<<1, 256, 0, stream>>>(pool, cnt, Wlin, blin, out);
}